// mLSTMLayer_28638841930504
// MI455X (gfx1250) — compile-verified
//
#include <hip/hip_runtime.h>
#include <math.h>

typedef float v2f __attribute__((ext_vector_type(2)));
typedef float v8f __attribute__((ext_vector_type(8)));

#define B_   2
#define S_   1024
#define D_   1024
#define NH_  8
#define DQK_ 64
#define DH_  128
#define M_   (B_*S_)          /* 2048 */
#define CAP_ 15.0f
#define EPS_ 1e-6f

#if defined(__gfx1250__) && __has_builtin(__builtin_amdgcn_global_load_async_to_lds_b128) && \
    __has_builtin(__builtin_amdgcn_s_wait_asynccnt)
#define USE_ASYNC_LDS 1
typedef int i32x4 __attribute__((ext_vector_type(4)));
typedef __attribute__((address_space(1))) i32x4 gv4i;   // global 16B vector
typedef __attribute__((address_space(3))) i32x4 lv4i;   // LDS 16B vector
#else
#define USE_ASYNC_LDS 0
#endif

// ---------------------------------------------------------------------------
// GEMM: C[M,N] = A[M,K] @ W[K,N], all f32, via V_WMMA_F32_16X16X4_F32.
// 256 threads = 8 waves (2 in M x 4 in N). Block tile 64(M) x 128(N).
// K staged in chunks of 32 through LDS (padded strides 36 / 132 floats),
// using GLOBAL_LOAD_ASYNC_TO_LDS_B128 when available (ASYNCcnt-tracked),
// plus global_prefetch of the next K-chunk tiles.
// Requires M%64==0, N%128==0, K%32==0 (true for all calls here).
// ---------------------------------------------------------------------------
__global__ __launch_bounds__(256) void gemm_f32_wmma(const float* __restrict__ A,
                                                     const float* __restrict__ Bm,
                                                     float* __restrict__ C,
                                                     int M, int N, int K) {
  __shared__ float As[64 * 36];
  __shared__ float Bs[32 * 132];
  const int tid  = threadIdx.x;
  const int w    = tid >> 5, lane = tid & 31;
  const int wm   = w >> 2,   wn   = w & 3;
  const int r    = lane & 15, kh  = lane >> 4;   // row/col within 16, K-half
  const int m0   = blockIdx.y * 64, n0 = blockIdx.x * 128;

  v8f acc00 = {}, acc01 = {}, acc10 = {}, acc11 = {};

  for (int k0 = 0; k0 < K; k0 += 32) {
#if USE_ASYNC_LDS
    // ---- async staging: memory pipe writes LDS directly, wave keeps issuing
    {
      int f4 = tid;
#pragma unroll
      for (int rep = 0; rep < 2; ++rep) {        // A tile 64x32
        const int row = f4 >> 3, c4 = (f4 & 7) << 2;
        __builtin_amdgcn_global_load_async_to_lds_b128(
            (gv4i*)&A[(size_t)(m0 + row) * K + k0 + c4],
            (lv4i*)&As[row * 36 + c4], 0, 0);
        f4 += 256;
      }
      f4 = tid;
#pragma unroll
      for (int rep = 0; rep < 4; ++rep) {        // B tile 32x128
        const int row = f4 >> 5, c4 = (f4 & 31) << 2;
        __builtin_amdgcn_global_load_async_to_lds_b128(
            (gv4i*)&Bm[(size_t)(k0 + row) * N + n0 + c4],
            (lv4i*)&Bs[row * 132 + c4], 0, 0);
        f4 += 256;
      }
      // pull next chunk toward this WGP while WMMAs run on the current one
      if (k0 + 32 < K) {
        __builtin_prefetch(&A[(size_t)(m0 + (tid >> 2)) * K + k0 + 32 + ((tid & 3) << 3)], 0, 3);
        __builtin_prefetch(&Bm[(size_t)(k0 + 32 + (tid >> 3)) * N + n0 + ((tid & 7) << 4)], 0, 3);
      }
      __builtin_amdgcn_s_wait_asynccnt(0);
    }
#else
    // ---- fallback staging: global -> VGPR -> ds_store
    {
      int f4 = tid;
#pragma unroll
      for (int rep = 0; rep < 2; ++rep) {        // A tile 64x32
        const int row = f4 >> 3, c4 = (f4 & 7) << 2;
        const float4 g = *reinterpret_cast<const float4*>(&A[(size_t)(m0 + row) * K + k0 + c4]);
        *reinterpret_cast<float4*>(&As[row * 36 + c4]) = g;
        f4 += 256;
      }
      f4 = tid;
#pragma unroll
      for (int rep = 0; rep < 4; ++rep) {        // B tile 32x128
        const int row = f4 >> 5, c4 = (f4 & 31) << 2;
        const float4 g = *reinterpret_cast<const float4*>(&Bm[(size_t)(k0 + row) * N + n0 + c4]);
        *reinterpret_cast<float4*>(&Bs[row * 132 + c4]) = g;
        f4 += 256;
      }
      if (k0 + 32 < K) {
        __builtin_prefetch(&A[(size_t)(m0 + (tid >> 2)) * K + k0 + 32 + ((tid & 3) << 3)], 0, 3);
        __builtin_prefetch(&Bm[(size_t)(k0 + 32 + (tid >> 3)) * N + n0 + ((tid & 7) << 4)], 0, 3);
      }
    }
#endif
    __syncthreads();
#pragma unroll
    for (int kk = 0; kk < 32; kk += 4) {
      const int kb = kk + 2 * kh;             // this lane's K pair base
      v2f a0, a1, b0, b1;
      a0[0] = As[(wm * 32 +      r) * 36 + kb];
      a0[1] = As[(wm * 32 +      r) * 36 + kb + 1];
      a1[0] = As[(wm * 32 + 16 + r) * 36 + kb];
      a1[1] = As[(wm * 32 + 16 + r) * 36 + kb + 1];
      const int cB = wn * 32 + r;
      b0[0] = Bs[(kb    ) * 132 + cB];
      b0[1] = Bs[(kb + 1) * 132 + cB];
      b1[0] = Bs[(kb    ) * 132 + cB + 16];
      b1[1] = Bs[(kb + 1) * 132 + cB + 16];
      acc00 = __builtin_amdgcn_wmma_f32_16x16x4_f32(false, a0, false, b0, (short)0, acc00, false, false);
      acc01 = __builtin_amdgcn_wmma_f32_16x16x4_f32(false, a0, false, b1, (short)0, acc01, false, false);
      acc10 = __builtin_amdgcn_wmma_f32_16x16x4_f32(false, a1, false, b0, (short)0, acc10, false, false);
      acc11 = __builtin_amdgcn_wmma_f32_16x16x4_f32(false, a1, false, b1, (short)0, acc11, false, false);
    }
    __syncthreads();
  }

  // C/D layout: VGPR j -> row (j + 8*kh), col = r
#pragma unroll
  for (int j = 0; j < 8; ++j) {
    const int row  = m0 + wm * 32 + j + 8 * kh;
    const int colb = n0 + wn * 32 + r;
    C[(size_t)row * N + colb]             = acc00[j];
    C[(size_t)row * N + colb + 16]        = acc01[j];
    C[(size_t)(row + 16) * N + colb]      = acc10[j];
    C[(size_t)(row + 16) * N + colb + 16] = acc11[j];
  }
}

// ---------------------------------------------------------------------------
// Gate projections (N = NH = 8, too narrow for WMMA): i = x@Wi + bi, f = x@Wf + bf
// ---------------------------------------------------------------------------
__global__ void gates_kernel(const float* __restrict__ x,
                             const float* __restrict__ Wi, const float* __restrict__ bi,
                             const float* __restrict__ Wf, const float* __restrict__ bf,
                             float* __restrict__ Ig, float* __restrict__ Fg) {
  const int idx = blockIdx.x * blockDim.x + threadIdx.x;
  if (idx >= M_ * NH_) return;
  const int m = idx / NH_, h = idx % NH_;
  const float* xr = &x[(size_t)m * D_];
  float si = 0.f, sf = 0.f;
  for (int k = 0; k < D_; ++k) {
    const float xv = xr[k];
    si = fmaf(xv, Wi[k * NH_ + h], si);
    sf = fmaf(xv, Wf[k * NH_ + h], sf);
  }
  Ig[idx] = si + bi[h];
  Fg[idx] = sf + bf[h];
}

// ---------------------------------------------------------------------------
// mLSTM scan: one 256-thread block per (b,h). State C[64][128] in registers:
// thread (dj = tid&127, half = tid>>7) owns C[32*half .. +31][dj] (32 regs).
// Per step: stabilized gates (replicated per thread), rank-1 state update,
// h_t[dj] = scale * sum_di C[di][dj] * q[di] (cross-half reduce via LDS).
// ---------------------------------------------------------------------------
__global__ __launch_bounds__(256) void scan_kernel(const float* __restrict__ Q,   // [M, NH*DQK]
                                                   const float* __restrict__ Kp,  // [M, NH*DQK]
                                                   const float* __restrict__ V,   // [M, NH*DH]
                                                   const float* __restrict__ Ig,  // [M, NH]
                                                   const float* __restrict__ Fg,  // [M, NH]
                                                   float* __restrict__ H) {       // [B][NH][S][DH]
  const int b = blockIdx.x / NH_, h = blockIdx.x % NH_;
  const int tid = threadIdx.x;
  const int dj = tid & 127, half = tid >> 7;
  __shared__ float sq[64], sk[64], sv[128], sp[256];

  float c[32];
#pragma unroll
  for (int rr = 0; rr < 32; ++rr) c[rr] = 0.f;
  float mrun = 0.f;
  const float scale = 0.125f;  // DQK^-0.5 = 64^-0.5

  for (int t = 0; t < S_; ++t) {
    const size_t m = (size_t)b * S_ + t;
    if (tid < 64) {
      sq[tid] = Q[m * (NH_ * DQK_) + h * DQK_ + tid];
      sk[tid] = Kp[m * (NH_ * DQK_) + h * DQK_ + tid];
    } else if (tid < 192) {
      sv[tid - 64] = V[m * (NH_ * DH_) + h * DH_ + (tid - 64)];
    }
    __syncthreads();

    const float it = Ig[m * NH_ + h];
    const float ft = Fg[m * NH_ + h];
    // log_sigmoid(ft), numerically stable
    const float flog = (ft >= 0.f) ? -log1pf(expf(-ft)) : (ft - log1pf(expf(ft)));
    const float mnew = fmaxf(flog + mrun, it);
    const float fact = expf(flog + mrun - mnew);
    const float iact = expf(it - mnew);
    mrun = mnew;

    const float vj = sv[dj] * iact;
    float partial = 0.f;
#pragma unroll
    for (int rr = 0; rr < 32; ++rr) {
      const int di = half * 32 + rr;
      c[rr] = fmaf(fact, c[rr], sk[di] * vj);      // C_new = f*C + i*(k v^T)
      partial = fmaf(c[rr], sq[di], partial);      // h partial = C_new^T q
    }
    sp[tid] = partial;
    __syncthreads();
    if (half == 0) {
      H[(((size_t)(b * NH_ + h)) * S_ + t) * DH_ + dj] = (sp[dj] + sp[dj + 128]) * scale;
    }
    // next-iter __syncthreads() (after staging) orders sp reads vs. rewrites
  }
}

// ---------------------------------------------------------------------------
// Per-head layernorm + output gate: one 128-thread block per (b,h,s).
// ---------------------------------------------------------------------------
__global__ __launch_bounds__(128) void lngate_kernel(const float* __restrict__ H,   // [B][NH][S][DH]
                                                     const float* __restrict__ OG,  // [M, D]
                                                     const float* __restrict__ lnw, // [NH,DH]
                                                     const float* __restrict__ lnb, // [NH,DH]
                                                     float* __restrict__ G) {       // [M, D]
  const int blk = blockIdx.x;            // (b*NH + h)*S + s
  const int s = blk % S_;
  const int h = (blk / S_) % NH_;
  const int b = blk / (S_ * NH_);
  const int dj = threadIdx.x;
  __shared__ float red[128];

  const float val = H[(size_t)blk * DH_ + dj];
  red[dj] = val;
  __syncthreads();
  for (int off = 64; off > 0; off >>= 1) {
    if (dj < off) red[dj] += red[dj + off];
    __syncthreads();
  }
  const float mean = red[0] * (1.f / 128.f);
  __syncthreads();
  const float dv = val - mean;
  red[dj] = dv * dv;
  __syncthreads();
  for (int off = 64; off > 0; off >>= 1) {
    if (dj < off) red[dj] += red[dj + off];
    __syncthreads();
  }
  const float var = red[0] * (1.f / 128.f);

  const float hn = dv * rsqrtf(var + EPS_) * lnw[h * DH_ + dj] + lnb[h * DH_ + dj];
  const size_t m = (size_t)b * S_ + s;
  const float og = OG[m * D_ + h * DH_ + dj];
  const float o  = 1.f / (1.f + expf(-tanhf(og * (1.f / CAP_)) * CAP_));
  G[m * D_ + h * DH_ + dj] = hn * o;
}

// ---------------------------------------------------------------------------
// Launch: projections (WMMA) -> gates -> scan -> LN*gate -> output GEMM (WMMA)
// ---------------------------------------------------------------------------
extern "C" void kernel_launch(void* const* d_in, const int* in_sizes, int n_in,
                              void* d_out, int out_size, void* d_ws, size_t ws_size,
                              hipStream_t stream) {
  const float* x    = (const float*)d_in[0];
  const float* Wq   = (const float*)d_in[1];
  const float* Wk   = (const float*)d_in[2];
  const float* Wv   = (const float*)d_in[3];
  const float* Wi   = (const float*)d_in[4];
  const float* bi   = (const float*)d_in[5];
  const float* Wf   = (const float*)d_in[6];
  const float* bf   = (const float*)d_in[7];
  const float* Wog  = (const float*)d_in[8];
  const float* lnw  = (const float*)d_in[9];
  const float* lnb  = (const float*)d_in[10];
  const float* Wout = (const float*)d_in[11];
  float* out = (float*)d_out;

  float* ws  = (float*)d_ws;
  float* Qp  = ws;                           // 2048*512
  float* Kp  = Qp  + (size_t)M_ * 512;       // 2048*512
  float* Vp  = Kp  + (size_t)M_ * 512;       // 2048*1024
  float* OGp = Vp  + (size_t)M_ * 1024;      // 2048*1024
  float* Ip  = OGp + (size_t)M_ * 1024;      // 2048*8
  float* Fp  = Ip  + (size_t)M_ * NH_;       // 2048*8
  float* Hp  = Fp  + (size_t)M_ * NH_;       // 2048*1024
  float* Gp  = Hp  + (size_t)M_ * 1024;      // 2048*1024

  const dim3 blk(256);
  gemm_f32_wmma<<<dim3(512 / 128,  M_ / 64), blk, 0, stream>>>(x,  Wq,   Qp,  M_, 512,  D_);
  gemm_f32_wmma<<<dim3(512 / 128,  M_ / 64), blk, 0, stream>>>(x,  Wk,   Kp,  M_, 512,  D_);
  gemm_f32_wmma<<<dim3(1024 / 128, M_ / 64), blk, 0, stream>>>(x,  Wv,   Vp,  M_, 1024, D_);
  gemm_f32_wmma<<<dim3(1024 / 128, M_ / 64), blk, 0, stream>>>(x,  Wog,  OGp, M_, 1024, D_);
  gates_kernel<<<(M_ * NH_ + 255) / 256, 256, 0, stream>>>(x, Wi, bi, Wf, bf, Ip, Fp);
  scan_kernel<<<B_ * NH_, 256, 0, stream>>>(Qp, Kp, Vp, Ip, Fp, Hp);
  lngate_kernel<<<B_ * NH_ * S_, 128, 0, stream>>>(Hp, OGp, lnw, lnb, Gp);
  gemm_f32_wmma<<<dim3(1024 / 128, M_ / 64), blk, 0, stream>>>(Gp, Wout, out, M_, 1024, D_);
}